// Masked_G_Attention_27487790695183
// MI455X (gfx1250) — compile-verified
//
#include <hip/hip_runtime.h>
#include <hip/hip_bf16.h>

// ---------------------------------------------------------------------------
// Masked G-Attention (causal linear attention) for MI455X / gfx1250 (CDNA5).
// fp32 end-to-end using V_WMMA_F32_16X16X4_F32 (wave32).
// Attention scan is a 3-phase split-scan:
//   A) per-segment K^T V sums (WMMA, fully parallel)
//   B) exclusive prefix over segments (tiny)
//   C) chunked in-segment scan seeded with the prefix state (parallel)
// ---------------------------------------------------------------------------

#define BATCH   2
#define NSEQ    2048
#define DIM     512
#define HEADS   8
#define DH      64
#define TRIPLE  (3 * DIM)
#define NSEG    16
#define SEGLEN  (NSEQ / NSEG)       // 128
#define EPSV    1e-7f

typedef __attribute__((ext_vector_type(2))) float v2f;
typedef __attribute__((ext_vector_type(8))) float v8f;

// D = A(16x4) * B(4x16) + C(16x16), all fp32.
// 8-arg VOP3P pattern: (neg_a, A, neg_b, B, c_mod, C, reuse_a, reuse_b)
static __device__ __forceinline__ v8f wmma_f32(v2f a, v2f b, v8f c) {
  return __builtin_amdgcn_wmma_f32_16x16x4_f32(
      false, a, false, b, (short)0, c, false, false);
}

static __device__ __forceinline__ v2f ld2(const float* p) {
  return *(const v2f*)p;   // 8B-aligned by construction (even float offsets)
}

// ---------------------------------------------------------------------------
// Kernel 1: qkv = x @ w_qkv^T, scattered into q/k/v with [b,h,n,d] layout.
// A-frag (16x4): lane L -> row L&15, k = 2*(L>>4)+{0,1}
// B-frag (4x16): lane L -> col L&15, k = 2*(L>>4)+{0,1}; B[k,n] = w[c0+n, k]
// C-frag: vgpr r, lane L -> (m = r + 8*(L>>4), n = L&15)
// ---------------------------------------------------------------------------
__global__ void __launch_bounds__(256)
qkv_gemm_kernel(const float* __restrict__ x, const float* __restrict__ w,
                float* __restrict__ qb, float* __restrict__ kb,
                float* __restrict__ vb) {
  const int lane = threadIdx.x & 31;
  const int wid  = threadIdx.x >> 5;
  const int gid  = blockIdx.x * 8 + wid;
  const int tr   = gid / (TRIPLE / 16);
  const int tc   = gid % (TRIPLE / 16);
  const int row0 = tr * 16, c0 = tc * 16;
  const int lp   = lane & 15;
  const int hl   = lane >> 4;
  const int h2   = hl * 2;

  v8f acc = {};
  const float* ap = x + (size_t)(row0 + lp) * DIM + h2;
  const float* bp = w + (size_t)(c0 + lp) * DIM + h2;
#pragma unroll 4
  for (int k0 = 0; k0 < DIM; k0 += 4)
    acc = wmma_f32(ld2(ap + k0), ld2(bp + k0), acc);

  const int c    = c0 + lp;
  const int mat  = c / DIM;        // 0=q 1=k 2=v (tile never straddles: 512%16==0)
  const int ci   = c % DIM;
  const int head = ci / DH, dcol = ci % DH;
  float* dst = (mat == 0) ? qb : ((mat == 1) ? kb : vb);
#pragma unroll
  for (int r = 0; r < 8; ++r) {
    const int R  = row0 + r + 8 * hl;
    const int bb = R >> 11, nn = R & (NSEQ - 1);
    dst[(((size_t)(bb * HEADS + head)) * NSEQ + nn) * DH + dcol] = acc[r];
  }
}

// ---------------------------------------------------------------------------
// Kernel 2: per-head LayerNorm over dim_head=64, in place on k and v.
// One wave32 per row; each lane owns elements {lane, lane+32}.
// ---------------------------------------------------------------------------
__global__ void __launch_bounds__(256)
ln_kernel(float* __restrict__ kb, float* __restrict__ vb,
          const float* __restrict__ lnkw, const float* __restrict__ lnkb,
          const float* __restrict__ lnvw, const float* __restrict__ lnvb) {
  const int lane = threadIdx.x & 31;
  const int wid  = threadIdx.x >> 5;
  const int row  = blockIdx.x * 8 + wid;          // 0 .. 2*B*H*N-1
  const int BHN  = BATCH * HEADS * NSEQ;
  const int which = (row >= BHN);
  const int rr    = which ? row - BHN : row;
  float* buf        = which ? vb   : kb;
  const float* wpar = which ? lnvw : lnkw;
  const float* bpar = which ? lnvb : lnkb;
  const int h = (rr / NSEQ) % HEADS;
  float* p = buf + (size_t)rr * DH;

  float x0 = p[lane], x1 = p[lane + 32];
  float s = x0 + x1, sq = x0 * x0 + x1 * x1;
#pragma unroll
  for (int m = 1; m < 32; m <<= 1) {
    s  += __shfl_xor(s,  m, 32);
    sq += __shfl_xor(sq, m, 32);
  }
  const float mu  = s * (1.f / 64.f);
  const float var = sq * (1.f / 64.f) - mu * mu;
  const float rs  = rsqrtf(var + EPSV);
  p[lane]      = (x0 - mu) * rs * wpar[h * DH + lane]      + bpar[h * DH + lane];
  p[lane + 32] = (x1 - mu) * rs * wpar[h * DH + lane + 32] + bpar[h * DH + lane + 32];
}

// ---------------------------------------------------------------------------
// Phase A: per-segment state sums.  grid = (NSEG, B*H), 4 waves/block.
// Wave w owns e-stripe [16w,16w+16): S_seg = K_seg^T V_seg via WMMA; wave 0
// also records the k column-sums of the segment.
// ---------------------------------------------------------------------------
__global__ void __launch_bounds__(128)
segstate_kernel(const float* __restrict__ kb, const float* __restrict__ vb,
                float* __restrict__ sseg, float* __restrict__ kcol) {
  const int lane = threadIdx.x & 31;
  const int wid  = threadIdx.x >> 5;
  const int lp   = lane & 15;
  const int hl   = lane >> 4;
  const int h2   = hl * 2;
  const int e0   = wid * 16;
  const int seg  = blockIdx.x;
  const int bh   = blockIdx.y;

  const float* kp = kb + ((size_t)bh * NSEQ + seg * SEGLEN) * DH;
  const float* vp = vb + ((size_t)bh * NSEQ + seg * SEGLEN) * DH;

  v8f S[4];
#pragma unroll
  for (int t = 0; t < 4; ++t) S[t] = (v8f){};
  float cs[4] = {0.f, 0.f, 0.f, 0.f};

  for (int j0 = 0; j0 < SEGLEN; j0 += 4) {
    const int r0 = j0 + h2;
    v2f bf;
    bf[0] = vp[(size_t)r0 * DH + e0 + lp];
    bf[1] = vp[(size_t)(r0 + 1) * DH + e0 + lp];
#pragma unroll
    for (int t = 0; t < 4; ++t) {
      v2f af;                                    // K^T A-frag
      af[0] = kp[(size_t)r0 * DH + 16 * t + lp];
      af[1] = kp[(size_t)(r0 + 1) * DH + 16 * t + lp];
      S[t] = wmma_f32(af, bf, S[t]);
      cs[t] += af[0] + af[1];
    }
  }

  float* sp = sseg + ((size_t)bh * NSEG + seg) * (DH * DH);
#pragma unroll
  for (int t = 0; t < 4; ++t)
#pragma unroll
    for (int r = 0; r < 8; ++r)
      sp[(16 * t + r + 8 * hl) * DH + e0 + lp] = S[t][r];
  if (wid == 0) {
#pragma unroll
    for (int t = 0; t < 4; ++t) {
      float c = cs[t] + __shfl_xor(cs[t], 16, 32);   // both row-halves
      if (hl == 0) kcol[((size_t)bh * NSEG + seg) * DH + 16 * t + lp] = c;
    }
  }
}

// ---------------------------------------------------------------------------
// Phase B: exclusive prefix over segments, in place.  grid = B*H blocks.
// Each thread owns 16 of the 4096 state elements; threads <64 also do kcol.
// ---------------------------------------------------------------------------
__global__ void __launch_bounds__(256)
segprefix_kernel(float* __restrict__ sseg, float* __restrict__ kcol) {
  const int bh  = blockIdx.x;
  const int tid = threadIdx.x;
  float* sp = sseg + (size_t)bh * NSEG * (DH * DH);
  float run[16];
#pragma unroll
  for (int j = 0; j < 16; ++j) run[j] = 0.f;
  for (int seg = 0; seg < NSEG; ++seg) {
    float* p = sp + (size_t)seg * (DH * DH);
#pragma unroll
    for (int j = 0; j < 16; ++j) {
      const int i = tid + 256 * j;
      const float tmp = p[i];
      p[i] = run[j];
      run[j] += tmp;
    }
  }
  if (tid < DH) {
    float* kp = kcol + (size_t)bh * NSEG * DH;
    float r2 = 0.f;
    for (int seg = 0; seg < NSEG; ++seg) {
      const float tmp = kp[seg * DH + tid];
      kp[seg * DH + tid] = r2;
      r2 += tmp;
    }
  }
}

// ---------------------------------------------------------------------------
// Phase C: chunked causal scan within one segment, seeded by prefix state.
// grid = (NSEG, B*H), 4 waves/block; wave w owns e-stripe [16w,16w+16).
// Per chunk (C=16 rows):
//   P = Q K^T (16 WMMAs) -> mask; row-sums give intra denominator
//   out = Q@S_prev (16) + P_mask@V (4)
//   S += K^T V (16) ; kcum += colsum(K)
//   out *= 1 / (n * ((kcum_incl+eps)·q))
// ---------------------------------------------------------------------------
__global__ void __launch_bounds__(128)
linattn_kernel(const float* __restrict__ qb, const float* __restrict__ kb,
               const float* __restrict__ vb,
               const float* __restrict__ sseg, const float* __restrict__ kcol,
               float* __restrict__ ob) {
  __shared__ float Sld[DH * DH];     // S[d][e], row stride DH
  __shared__ float kcum[DH];         // running column-sum of k (global prefix)
  __shared__ float Pld[4][16 * 16];  // per-wave masked P (C/D -> A layout hop)
  __shared__ float denL[4][16];      // per-wave inter-denominator per row

  const int lane = threadIdx.x & 31;
  const int wid  = threadIdx.x >> 5;
  const int lp   = lane & 15;
  const int hl   = lane >> 4;
  const int h2   = hl * 2;
  const int e0   = wid * 16;
  const int seg  = blockIdx.x;
  const int bh   = blockIdx.y;
  const int b    = bh >> 3, h = bh & 7;

  const float* qp = qb + (size_t)bh * NSEQ * DH;
  const float* kp = kb + (size_t)bh * NSEQ * DH;
  const float* vp = vb + (size_t)bh * NSEQ * DH;
  const float* sp = sseg + ((size_t)bh * NSEG + seg) * (DH * DH);
  const float* cp = kcol + ((size_t)bh * NSEG + seg) * DH;
  float* op = ob + (size_t)b * NSEQ * DIM + h * DH;

  // seed with exclusive-prefix state
  for (int i = threadIdx.x; i < DH * DH; i += 128) Sld[i] = sp[i];
  for (int i = threadIdx.x; i < DH; i += 128) kcum[i] = cp[i];
  v8f S[4];
#pragma unroll
  for (int t = 0; t < 4; ++t)
#pragma unroll
    for (int r = 0; r < 8; ++r)
      S[t][r] = sp[(16 * t + r + 8 * hl) * DH + e0 + lp];
  __syncthreads();

  const float inv_n = 1.0f / (float)NSEQ;
  const int nend = (seg + 1) * SEGLEN;

  for (int n0 = seg * SEGLEN; n0 < nend; n0 += 16) {
    // ---- Q fragments for this chunk (A layout, all 16 K-steps) ----
    v2f qa[16];
    const float* qrow = qp + (size_t)(n0 + lp) * DH + h2;
#pragma unroll
    for (int s = 0; s < 16; ++s) qa[s] = ld2(qrow + 4 * s);

    // ---- P = Q K^T  (B-frag of K^T == A-frag addressing of K) ----
    v8f P = {};
    const float* krow = kp + (size_t)(n0 + lp) * DH + h2;
#pragma unroll
    for (int s = 0; s < 16; ++s)
      P = wmma_f32(qa[s], ld2(krow + 4 * s), P);

    // ---- causal mask + intra-denominator row sums ----
    float rsum[8];
#pragma unroll
    for (int r = 0; r < 8; ++r) {
      float pv = (lp <= r + 8 * hl) ? P[r] : 0.f;
      P[r] = pv;
      float t = pv;
#pragma unroll
      for (int m = 1; m < 16; m <<= 1) t += __shfl_xor(t, m, 32);  // over n
      rsum[r] = t;   // rowsum for m = r + 8*hl
    }

    // ---- inter-denominator: q_i . (kcum_prev + eps) ----
    float qdot = 0.f;
#pragma unroll
    for (int s = 0; s < 16; ++s) {
      const int d0 = 4 * s + h2;
      qdot += qa[s][0] * (kcum[d0] + EPSV) + qa[s][1] * (kcum[d0 + 1] + EPSV);
    }
    qdot += __shfl_xor(qdot, 16, 32);   // combine K halves; row = lane&15

    // ---- out_inter = Q @ S_prev (S in LDS, B-frag layout) ----
    v8f out = {};
#pragma unroll
    for (int s = 0; s < 16; ++s) {
      const int d0 = 4 * s + h2;
      v2f bf;
      bf[0] = Sld[d0 * DH + e0 + lp];
      bf[1] = Sld[(d0 + 1) * DH + e0 + lp];
      out = wmma_f32(qa[s], bf, out);
    }

    // ---- publish masked P (for A-layout reuse) and inter-denominator ----
#pragma unroll
    for (int r = 0; r < 8; ++r) Pld[wid][(r + 8 * hl) * 16 + lp] = P[r];
    if (hl == 0) denL[wid][lp] = qdot;
    __syncthreads();

    // ---- out_intra = P_mask @ V ----
#pragma unroll
    for (int s = 0; s < 4; ++s) {
      const int j0 = 4 * s + h2;
      v2f af = ld2(&Pld[wid][lp * 16 + j0]);
      v2f bf;
      bf[0] = vp[(size_t)(n0 + j0) * DH + e0 + lp];
      bf[1] = vp[(size_t)(n0 + j0 + 1) * DH + e0 + lp];
      out = wmma_f32(af, bf, out);
    }

    // ---- per-row 1/(n*den) ----
    float dinv[8];
#pragma unroll
    for (int r = 0; r < 8; ++r)
      dinv[r] = inv_n / (denL[wid][r + 8 * hl] + rsum[r]);

    // ---- state update S += K^T V ; wave0 accumulates kcum colsums ----
    float cs[4];
#pragma unroll
    for (int t = 0; t < 4; ++t) {
      float c = 0.f;
#pragma unroll
      for (int s = 0; s < 4; ++s) {
        const int j0 = 4 * s + h2;
        v2f af, bf;
        af[0] = kp[(size_t)(n0 + j0) * DH + 16 * t + lp];       // K^T A-frag
        af[1] = kp[(size_t)(n0 + j0 + 1) * DH + 16 * t + lp];
        bf[0] = vp[(size_t)(n0 + j0) * DH + e0 + lp];           // V B-frag
        bf[1] = vp[(size_t)(n0 + j0 + 1) * DH + e0 + lp];
        S[t] = wmma_f32(af, bf, S[t]);
        c += af[0] + af[1];
      }
      cs[t] = c;
    }
#pragma unroll
    for (int t = 0; t < 4; ++t)
#pragma unroll
      for (int r = 0; r < 8; ++r)
        Sld[(16 * t + r + 8 * hl) * DH + e0 + lp] = S[t][r];
    if (wid == 0) {
#pragma unroll
      for (int t = 0; t < 4; ++t) {
        float c = cs[t] + __shfl_xor(cs[t], 16, 32);  // both row-halves, d=16t+lp
        if (hl == 0) kcum[16 * t + lp] += c;
      }
    }

    // ---- write scaled output [b, n, h*64 + e] ----
#pragma unroll
    for (int r = 0; r < 8; ++r)
      op[(size_t)(n0 + r + 8 * hl) * DIM + e0 + lp] = out[r] * dinv[r];

    __syncthreads();   // S/kcum consistent before next chunk
  }
}

// ---------------------------------------------------------------------------
// Kernel 4: out = attn @ w_out^T + b_out  (same WMMA tiling as kernel 1)
// ---------------------------------------------------------------------------
__global__ void __launch_bounds__(256)
outproj_kernel(const float* __restrict__ a, const float* __restrict__ w,
               const float* __restrict__ bias, float* __restrict__ out) {
  const int lane = threadIdx.x & 31;
  const int wid  = threadIdx.x >> 5;
  const int gid  = blockIdx.x * 8 + wid;
  const int tr   = gid >> 5;           // 512/16 = 32 column tiles
  const int tc   = gid & 31;
  const int row0 = tr * 16, c0 = tc * 16;
  const int lp   = lane & 15;
  const int hl   = lane >> 4;
  const int h2   = hl * 2;

  v8f acc = {};
  const float* ap = a + (size_t)(row0 + lp) * DIM + h2;
  const float* bp = w + (size_t)(c0 + lp) * DIM + h2;
#pragma unroll 4
  for (int k0 = 0; k0 < DIM; k0 += 4)
    acc = wmma_f32(ld2(ap + k0), ld2(bp + k0), acc);

  const float bv = bias[c0 + lp];
#pragma unroll
  for (int r = 0; r < 8; ++r)
    out[(size_t)(row0 + r + 8 * hl) * DIM + c0 + lp] = acc[r] + bv;
}

// ---------------------------------------------------------------------------
extern "C" void kernel_launch(void* const* d_in, const int* in_sizes, int n_in,
                              void* d_out, int out_size, void* d_ws, size_t ws_size,
                              hipStream_t stream) {
  const float* x     = (const float*)d_in[0];
  const float* w_qkv = (const float*)d_in[1];
  const float* lnk_w = (const float*)d_in[2];
  const float* lnk_b = (const float*)d_in[3];
  const float* lnv_w = (const float*)d_in[4];
  const float* lnv_b = (const float*)d_in[5];
  const float* w_out = (const float*)d_in[6];
  const float* b_out = (const float*)d_in[7];
  float* out = (float*)d_out;

  float* ws = (float*)d_ws;
  const size_t SEG = (size_t)BATCH * HEADS * NSEQ * DH;      // 2,097,152 floats
  const int    BH  = BATCH * HEADS;                          // 16
  float* qb   = ws;
  float* kb   = ws + SEG;
  float* vb   = ws + 2 * SEG;
  float* ab   = ws + 3 * SEG;                                // [b, n, h*d]
  float* sseg = ws + 4 * SEG;                                // BH*NSEG*64*64
  float* kcol = sseg + (size_t)BH * NSEG * DH * DH;          // BH*NSEG*64

  // 1) qkv projection: 4096/16 x 1536/16 = 24576 tiles, 8 waves/block
  qkv_gemm_kernel<<<24576 / 8, 256, 0, stream>>>(x, w_qkv, qb, kb, vb);

  // 2) per-head LN on k and v: 2*B*H*N = 65536 rows, 8 waves/block
  ln_kernel<<<65536 / 8, 256, 0, stream>>>(kb, vb, lnk_w, lnk_b, lnv_w, lnv_b);

  // 3a) per-segment K^T V sums (parallel over 256 blocks)
  segstate_kernel<<<dim3(NSEG, BH), 128, 0, stream>>>(kb, vb, sseg, kcol);

  // 3b) exclusive prefix over segments (16 blocks)
  segprefix_kernel<<<BH, 256, 0, stream>>>(sseg, kcol);

  // 3c) in-segment chunked scan, seeded by prefix state (256 blocks)
  linattn_kernel<<<dim3(NSEG, BH), 128, 0, stream>>>(qb, kb, vb, sseg, kcol, ab);

  // 4) output projection: 4096/16 x 512/16 = 8192 tiles, 8 waves/block
  outproj_kernel<<<8192 / 8, 256, 0, stream>>>(ab, w_out, b_out, out);
}